// VectorQuantizer_9689446220466
// MI455X (gfx1250) — compile-verified
//
#include <hip/hip_runtime.h>

typedef __attribute__((ext_vector_type(2))) float v2f;
typedef __attribute__((ext_vector_type(8))) float v8f;

#define G_ 64
#define K_ 256
#define D_ 8
#define N_ 8192
#define BPG 16                         // blocks per group
#define TOK_PER_BLOCK (N_ / BPG)       // 512 tokens per block
#define WAVES 8
#define TILES_PER_WAVE (TOK_PER_BLOCK / (WAVES * 16))  // 4
#define CBS 12                         // padded LDS row stride (floats): bank-conflict-free + 16B aligned

__global__ __launch_bounds__(256) void vq_wmma_kernel(
    const float* __restrict__ x,
    const float* __restrict__ codebook,
    const float* __restrict__ scale_p,
    float* __restrict__ out)
{
    __shared__ __align__(16) float cbLds[K_ * CBS];   // 12 KB scaled codebook
    __shared__ float c2Lds[K_];                       // 1 KB squared norms
    __shared__ int   idxBuf[WAVES][16];               // per-wave winners

    const int tid  = threadIdx.x;
    const int g    = blockIdx.x >> 4;                 // / BPG
    const int blkg = blockIdx.x & (BPG - 1);
    const float scale = scale_p[0];

    // Stage scaled codebook + norms into LDS: one code row per thread.
    {
        const int code = tid;                         // 256 threads == K_
        const float* src = codebook + ((size_t)g * K_ + code) * D_;
        float c2 = 0.f;
        #pragma unroll
        for (int i = 0; i < D_; ++i) {
            float v = src[i] * scale;
            cbLds[code * CBS + i] = v;
            c2 = __builtin_fmaf(v, v, c2);
        }
        c2Lds[code] = c2;
    }
    __syncthreads();

    const int wave  = tid >> 5;
    const int lane  = tid & 31;
    const int mrow  = lane & 15;                // token row / code column within tile
    const int dbase = (lane < 16) ? 0 : 2;      // K-slot base (ISA 16x4 fp32 A/B layout)
    const bool lo16 = (lane < 16);

    // Third A chunk: effective row [1,0,0,0] for every token -> adds B2 row0 (= c2) to acc.
    const v2f A2 = { lo16 ? 1.0f : 0.0f, 0.0f };

    // Preload code-tiles in WMMA-B layout. B0/B1 pre-scaled by -2 so the
    // accumulator directly produces c2 - 2*x.c (argmin-equivalent distance).
    v2f Breg0[16], Breg1[16], Breg2[16];
    #pragma unroll
    for (int t = 0; t < 16; ++t) {
        const int n = t * 16 + mrow;
        v2f b0 = *(const v2f*)&cbLds[n * CBS + dbase];        // d = dbase, dbase+1
        v2f b1 = *(const v2f*)&cbLds[n * CBS + 4 + dbase];    // d = 4+dbase, 5+dbase
        Breg0[t].x = -2.0f * b0.x;  Breg0[t].y = -2.0f * b0.y;
        Breg1[t].x = -2.0f * b1.x;  Breg1[t].y = -2.0f * b1.y;
        const float c2 = c2Lds[n];
        Breg2[t].x = lo16 ? c2 : 0.0f;                        // row K=0 carries c2[n]
        Breg2[t].y = 0.0f;                                    // unused rows finite zero
    }

    const size_t gBase   = (size_t)g * N_;
    const int    tokWave = blkg * TOK_PER_BLOCK + wave * (TILES_PER_WAVE * 16);

    for (int tile = 0; tile < TILES_PER_WAVE; ++tile) {
        const int tokBase = tokWave + tile * 16;

        // A-matrix: x tile, 16 tokens x 4 dims per chunk (lane=token, VGPR=K slot)
        const float* xp = x + (gBase + tokBase + mrow) * D_ + dbase;
        v2f a0 = *(const v2f*)(xp);
        v2f a1 = *(const v2f*)(xp + 4);

        // Packed (distance | code) in one float: low 8 mantissa bits carry the index.
        float best[8];
        #pragma unroll
        for (int v = 0; v < 8; ++v) best[v] = __uint_as_float(0x7F7FFFFFu);

        #pragma unroll
        for (int t = 0; t < 16; ++t) {
            v8f c = {};
            c = __builtin_amdgcn_wmma_f32_16x16x4_f32(false, a0, false, Breg0[t],
                                                      (short)0, c, false, false);
            c = __builtin_amdgcn_wmma_f32_16x16x4_f32(false, a1, false, Breg1[t],
                                                      (short)0, c, false, false);
            c = __builtin_amdgcn_wmma_f32_16x16x4_f32(false, A2, false, Breg2[t],
                                                      (short)0, c, false, false);
            const unsigned n = (unsigned)(t * 16 + mrow);
            #pragma unroll
            for (int v = 0; v < 8; ++v) {
                unsigned b = (__float_as_uint(c[v]) & 0xFFFFFF00u) | n;  // v_and_or_b32
                best[v] = fminf(best[v], __uint_as_float(b));            // v_min(3)_num_f32
            }
        }

        // Butterfly min-reduce within each 16-lane half (codes live in lane%16).
        #pragma unroll
        for (int m = 1; m <= 8; m <<= 1) {
            #pragma unroll
            for (int v = 0; v < 8; ++v) {
                best[v] = fminf(best[v], __shfl_xor(best[v], m, 32));
            }
        }

        // Lanes 0 and 16 publish winners: rows v hold tokens v (half 0) / v+8 (half 1).
        if (mrow == 0) {
            const int half = (lane >> 4) * 8;
            #pragma unroll
            for (int v = 0; v < 8; ++v)
                idxBuf[wave][half + v] = (int)(__float_as_uint(best[v]) & 255u);
        }
        __syncthreads();

        // Gather + store: 2 lanes per token, one float4 each (coalesced 512B/wave).
        {
            const int tt   = lane >> 1;
            const int comp = (lane & 1) * 4;
            const int idx  = idxBuf[wave][tt];
            const float4 val = *(const float4*)&cbLds[idx * CBS + comp];
            *(float4*)&out[(gBase + tokBase + tt) * D_ + comp] = val;
        }
        __syncthreads();
    }
}

extern "C" void kernel_launch(void* const* d_in, const int* in_sizes, int n_in,
                              void* d_out, int out_size, void* d_ws, size_t ws_size,
                              hipStream_t stream) {
    const float* x        = (const float*)d_in[0];
    const float* codebook = (const float*)d_in[1];
    const float* scale    = (const float*)d_in[2];
    float* out            = (float*)d_out;

    dim3 grid(G_ * BPG);
    dim3 block(256);
    vq_wmma_kernel<<<grid, block, 0, stream>>>(x, codebook, scale, out);
}